// YOLO3_16140487098776
// MI455X (gfx1250) — compile-verified
//
#include <hip/hip_runtime.h>
#include <hip/hip_bf16.h>

#define N_ANCH   64512
#define NCLS     80
#define NCOL     84          // NUM_CLASSES + 4
#define TOPK_N   1024
#define MAXOUT   100
#define NEGV     (-1000000000.0f)
#define SCORE_T  0.5f

typedef __attribute__((ext_vector_type(2))) float v2f;
typedef __attribute__((ext_vector_type(8))) float v8f;

// Order-preserving float -> uint key (handles NEG padding naturally).
__device__ __forceinline__ unsigned scoreKey(const float* __restrict__ cla_reg,
                                             const float* __restrict__ obj,
                                             int a, int c) {
    float s = cla_reg[(size_t)a * NCOL + c] * obj[a];
    s = (s > SCORE_T) ? s : NEGV;
    unsigned u = __float_as_uint(s);
    return (u & 0x80000000u) ? ~u : (u | 0x80000000u);
}

// ---------------------------------------------------------------- decode boxes
__global__ void decode_kernel(const float* __restrict__ cla_reg,
                              const float* __restrict__ anchors,
                              float* __restrict__ boxes) {
    int a = blockIdx.x * 256 + threadIdx.x;
    if (a >= N_ANCH) return;
    const float* r = cla_reg + (size_t)a * NCOL + NCLS;
    float ry = 1.0f / (1.0f + __expf(-r[0]));
    float rx = 1.0f / (1.0f + __expf(-r[1]));
    float cy = anchors[a * 4 + 0] + ry;
    float cx = anchors[a * 4 + 1] + rx;
    float h  = anchors[a * 4 + 2] * __expf(r[2]);
    float w  = anchors[a * 4 + 3] * __expf(r[3]);
    boxes[a * 4 + 0] = cy - 0.5f * h;
    boxes[a * 4 + 1] = cx - 0.5f * w;
    boxes[a * 4 + 2] = cy + 0.5f * h;
    boxes[a * 4 + 3] = cx + 0.5f * w;
}

// ------------------------------------------------------- per-class top-1024
__global__ __launch_bounds__(1024) void topk_kernel(const float* __restrict__ cla_reg,
                                                    const float* __restrict__ obj,
                                                    const float* __restrict__ boxes,
                                                    float* __restrict__ topv,
                                                    float* __restrict__ topb) {
    const int c   = blockIdx.x;
    const int tid = threadIdx.x;
    __shared__ unsigned skey[TOPK_N];
    __shared__ int      sidx[TOPK_N];
    __shared__ int      scnt;
    __shared__ int      gcnt;

    // Binary search: largest pivot p with count(key >= p) >= TOPK_N.
    unsigned lo = 0u, hi = 0xFFFFFFFFu;
    for (int it = 0; it < 32; ++it) {
        if (lo >= hi) break;                       // uniform across block
        unsigned span = hi - lo;
        unsigned mid  = lo + (span >> 1) + (span & 1u);   // upper mid, > lo
        if (tid == 0) scnt = 0;
        __syncthreads();
        int local = 0;
        for (int a = tid; a < N_ANCH; a += 1024)
            local += (scoreKey(cla_reg, obj, a, c) >= mid) ? 1 : 0;
        atomicAdd(&scnt, local);
        __syncthreads();
        int cnt = scnt;
        __syncthreads();
        if (cnt >= TOPK_N) lo = mid; else hi = mid - 1;
    }
    const unsigned p = lo;

    // Gather: strictly-greater first, then pivot-equal fills the remainder.
    if (tid == 0) gcnt = 0;
    __syncthreads();
    for (int a = tid; a < N_ANCH; a += 1024) {
        unsigned k = scoreKey(cla_reg, obj, a, c);
        if (k > p) {
            int pos = atomicAdd(&gcnt, 1);
            if (pos < TOPK_N) { skey[pos] = k; sidx[pos] = a; }
        }
    }
    __syncthreads();
    for (int a = tid; a < N_ANCH; a += 1024) {
        unsigned k = scoreKey(cla_reg, obj, a, c);
        if (k == p) {
            int pos = atomicAdd(&gcnt, 1);
            if (pos < TOPK_N) { skey[pos] = k; sidx[pos] = a; }
        }
    }
    __syncthreads();

    // Bitonic sort, descending by key. 1024 threads / 1024 elements.
    for (int k = 2; k <= TOPK_N; k <<= 1) {
        for (int j = k >> 1; j > 0; j >>= 1) {
            __syncthreads();
            int ixj = tid ^ j;
            if (ixj > tid) {
                unsigned ka = skey[tid], kb = skey[ixj];
                bool desc = ((tid & k) == 0);
                if ((ka < kb) == desc) {
                    skey[tid] = kb; skey[ixj] = ka;
                    int ia = sidx[tid]; sidx[tid] = sidx[ixj]; sidx[ixj] = ia;
                }
            }
        }
    }
    __syncthreads();

    // Emit sorted scores + gathered boxes.
    unsigned k = skey[tid];
    float s = (k & 0x80000000u) ? __uint_as_float(k ^ 0x80000000u)
                                : __uint_as_float(~k);
    topv[(size_t)c * TOPK_N + tid] = s;
    int id = sidx[tid];
#pragma unroll
    for (int q = 0; q < 4; ++q)
        topb[((size_t)c * TOPK_N + tid) * 4 + q] = boxes[(size_t)id * 4 + q];
}

// --------------------------------------- suppression bitmask matrix via WMMA
// One wave handles a 16-row x 32-col tile. The union term area_i + area_j is
// computed as a rank-2 outer product with V_WMMA_F32_16X16X4_F32:
//   A(16x4) rows = [area_i, 1, 0, 0],  B(4x16) = [[1..1],[area_j],[0],[0]]
__global__ __launch_bounds__(256) void ioumask_kernel(const float* __restrict__ topb,
                                                      unsigned* __restrict__ masks) {
    const int c    = blockIdx.z;
    const int wave = threadIdx.x >> 5;
    const int lane = threadIdx.x & 31;
    const int i0   = (blockIdx.x * 8 + wave) * 16;
    const int j0   = blockIdx.y * 32;

    __shared__ float rb[8][16][4];
    __shared__ float cb[32][4];
    const float* tb = topb + (size_t)c * TOPK_N * 4;
    if (lane < 16) {
#pragma unroll
        for (int k = 0; k < 4; ++k) rb[wave][lane][k] = tb[(size_t)(i0 + lane) * 4 + k];
    }
    if (wave == 0) {
#pragma unroll
        for (int k = 0; k < 4; ++k) cb[lane][k] = tb[(size_t)(j0 + lane) * 4 + k];
    }
    __syncthreads();

    const int hsel = lane >> 4;   // 0: lanes 0-15 (K0/K1), 1: lanes 16-31 (K2/K3 = zero)
    const int l    = lane & 15;

    float areaI = (rb[wave][l][2] - rb[wave][l][0]) * (rb[wave][l][3] - rb[wave][l][1]);
    v2f A; A.x = hsel ? 0.0f : areaI; A.y = hsel ? 0.0f : 1.0f;

    unsigned b0[8], b1[8];
#pragma unroll
    for (int sub = 0; sub < 2; ++sub) {
        const int jb = sub * 16;
        float jy1 = cb[jb + l][0], jx1 = cb[jb + l][1];
        float jy2 = cb[jb + l][2], jx2 = cb[jb + l][3];
        float areaJ = (jy2 - jy1) * (jx2 - jx1);
        v2f B; B.x = hsel ? 0.0f : 1.0f; B.y = hsel ? 0.0f : areaJ;

        v8f acc = {};
        // D[i][j] = area_i + area_j   (union term, pre-intersection)
        acc = __builtin_amdgcn_wmma_f32_16x16x4_f32(false, A, false, B,
                                                    (short)0, acc, false, false);
#pragma unroll
        for (int v = 0; v < 8; ++v) {
            int m  = v + 8 * hsel;                 // C/D layout: lanes16-31 hold M+8
            int gi = i0 + m;
            int gj = j0 + jb + l;
            float iy1 = fmaxf(rb[wave][m][0], jy1);
            float ix1 = fmaxf(rb[wave][m][1], jx1);
            float iy2 = fminf(rb[wave][m][2], jy2);
            float ix2 = fminf(rb[wave][m][3], jx2);
            float inter = fmaxf(iy2 - iy1, 0.0f) * fmaxf(ix2 - ix1, 0.0f);
            // iou > 0.45  <=>  inter/(sum - inter + eps) > 0.45
            bool pred = (1.45f * inter > 0.45f * (acc[v] + 1e-9f)) && (gj > gi);
            unsigned bal = (unsigned)__ballot(pred);   // wave32: [15:0]=row m, [31:16]=row m+8
            if (sub == 0) b0[v] = bal; else b1[v] = bal;
        }
    }

    unsigned* mrow = masks + (size_t)c * TOPK_N * 32 + (j0 >> 5);
#pragma unroll
    for (int v = 0; v < 8; ++v) {
        if (lane == v)
            mrow[(size_t)(i0 + v) * 32]     = (b0[v] & 0xFFFFu) | (b1[v] << 16);
        if (lane == v + 8)
            mrow[(size_t)(i0 + v + 8) * 32] = (b0[v] >> 16) | (b1[v] & 0xFFFF0000u);
    }
}

// ------------------------------------------------------ serial NMS + compact
__global__ __launch_bounds__(256) void nms_kernel(const float* __restrict__ topv,
                                                  const float* __restrict__ topb,
                                                  const unsigned* __restrict__ masks,
                                                  float* __restrict__ out) {
    const int c   = blockIdx.x;
    const int tid = threadIdx.x;
    __shared__ unsigned keep[32];
    __shared__ float    sv[TOPK_N];
    __shared__ int      woff[32];

    if (tid < 32) keep[tid] = 0u;
    __syncthreads();
    for (int i = tid; i < TOPK_N; i += 256) {
        float s = topv[(size_t)c * TOPK_N + i];
        sv[i] = s;
        if (s > NEGV * 0.5f) atomicOr(&keep[i >> 5], 1u << (i & 31));
    }
    const unsigned* mbase = masks + (size_t)c * TOPK_N * 32;

    for (int i = 0; i < TOPK_N; ++i) {
        __syncthreads();
        bool kb = (keep[i >> 5] >> (i & 31)) & 1u;
        if (tid < 32 && (i + 1) < TOPK_N)
            __builtin_prefetch(&mbase[(size_t)(i + 1) * 32 + tid], 0, 0);
        if (kb && tid < 32)
            keep[tid] &= ~mbase[(size_t)i * 32 + tid];
    }
    __syncthreads();

    if (tid < 32) woff[tid] = __popc(keep[tid]);
    __syncthreads();
    if (tid == 0) {
        int acc = 0;
        for (int w = 0; w < 32; ++w) { int t = woff[w]; woff[w] = acc; acc += t; }
    }
    __syncthreads();

    float* out_cls = out;
    float* out_s   = out + 1 * NCLS * MAXOUT;
    float* out_b   = out + 2 * NCLS * MAXOUT;
    float* out_v   = out + 6 * NCLS * MAXOUT;
    for (int i = tid; i < TOPK_N; i += 256) {
        unsigned w = keep[i >> 5];
        if ((w >> (i & 31)) & 1u) {
            unsigned below = (i & 31) ? (w & ((1u << (i & 31)) - 1u)) : 0u;
            int rank = woff[i >> 5] + __popc(below);
            if (rank < MAXOUT) {
                int o = c * MAXOUT + rank;
                out_cls[o] = (float)c;
                out_s[o]   = sv[i];
                out_v[o]   = 1.0f;
#pragma unroll
                for (int k = 0; k < 4; ++k)
                    out_b[(size_t)o * 4 + k] = topb[((size_t)c * TOPK_N + i) * 4 + k];
            }
        }
    }
}

__global__ void zero_kernel(float* __restrict__ out, int n) {
    int i = blockIdx.x * 256 + threadIdx.x;
    if (i < n) out[i] = 0.0f;
}

// -------------------------------------------------------------------- launch
extern "C" void kernel_launch(void* const* d_in, const int* in_sizes, int n_in,
                              void* d_out, int out_size, void* d_ws, size_t ws_size,
                              hipStream_t stream) {
    const float* cla_reg = (const float*)d_in[0];
    const float* obj     = (const float*)d_in[1];
    const float* anchors = (const float*)d_in[2];
    float* out = (float*)d_out;
    char*  ws  = (char*)d_ws;

    // ws layout (bytes): boxes @0 (1.0MB), topv @1MB (320KB),
    // topb @1.5MB (1.25MB), masks @3MB (10MB). Total ~13MB.
    float*    boxes = (float*)(ws);
    float*    topv  = (float*)(ws + ((size_t)1 << 20));
    float*    topb  = (float*)(ws + ((size_t)3 << 19));
    unsigned* masks = (unsigned*)(ws + ((size_t)3 << 20));

    const int out_n = 6 * NCLS * MAXOUT + NCLS * MAXOUT;  // 56000
    zero_kernel<<<(out_n + 255) / 256, 256, 0, stream>>>(out, out_n);
    decode_kernel<<<(N_ANCH + 255) / 256, 256, 0, stream>>>(cla_reg, anchors, boxes);
    topk_kernel<<<NCLS, 1024, 0, stream>>>(cla_reg, obj, boxes, topv, topb);
    ioumask_kernel<<<dim3(8, 32, NCLS), 256, 0, stream>>>(topb, masks);
    nms_kernel<<<NCLS, 256, 0, stream>>>(topv, topb, masks, out);
}